// HyperbolicGCN_74363063763646
// MI455X (gfx1250) — compile-verified
//
#include <hip/hip_runtime.h>
#include <hip/hip_bf16.h>
#include <math.h>

#define EPSF  1e-7f
#define MINNF 1e-15f

typedef float v2f __attribute__((ext_vector_type(2)));
typedef float v8f __attribute__((ext_vector_type(8)));

static __device__ __forceinline__ float blk_reduce64(float v, volatile float* red) {
  const int t = threadIdx.x;
  red[t] = v;               __syncthreads();
  if (t < 32) red[t] += red[t + 32]; __syncthreads();
  if (t < 16) red[t] += red[t + 16]; __syncthreads();
  if (t < 8)  red[t] += red[t + 8];  __syncthreads();
  if (t < 4)  red[t] += red[t + 4];  __syncthreads();
  if (t < 2)  red[t] += red[t + 2];  __syncthreads();
  if (t < 1)  red[t] += red[t + 1];  __syncthreads();
  float r = red[0];         __syncthreads();
  return r;
}

// ---------------------------------------------------------------------------
// Kernel 1: adjacency scatter  adj[src,dst] += 1.0   (counts are integers ->
// float atomicAdd is exactly associative -> deterministic)
// ---------------------------------------------------------------------------
__global__ void hgcn_edge_scatter(const long long* __restrict__ ei,
                                  float* __restrict__ adj, int E) {
  int e = blockIdx.x * blockDim.x + threadIdx.x;
  if (e < E) {
    int s = (int)ei[e];
    int d = (int)ei[E + e];
    atomicAdd(&adj[s * 1024 + d], 1.0f);
  }
}

// ---------------------------------------------------------------------------
// Kernel 2: per-node phase A.
//  u      = logmap(o_in, x)          (time comp 0, spatial scaled)
//  mu     = u @ W^T                  (skip d=0, u0==0)
//  h      = hproj(expmap(o_out, proj_tan(mu)))
//  pt     = ptransp(o_out, h, b);  h = expmap(h, pt)
//  h_tan  = logmap(o_out, h);  s_l/s_r = h_tan . attn_w halves
//  q_i    = mdot(h,h)
// Block = 64 threads (one per output feature), one node per block.
// ---------------------------------------------------------------------------
__global__ __launch_bounds__(64)
void hgcn_nodeA(const float* __restrict__ x, const float* __restrict__ W,
                const float* __restrict__ b, const float* __restrict__ aw,
                float* __restrict__ H, float* __restrict__ Q,
                float* __restrict__ SL, float* __restrict__ SR) {
  __shared__ float xs[128];
  __shared__ float red[64];
  const int i = blockIdx.x;
  const int k = threadIdx.x;

  xs[k]      = x[i * 128 + k];
  xs[k + 64] = x[i * 128 + 64 + k];
  __syncthreads();

  // ---- logmap(o_in, x): scale for spatial components -----------------------
  float v = ((k == 0) ? 0.f : xs[k] * xs[k]) + xs[k + 64] * xs[k + 64];
  float sx = blk_reduce64(v, red);                       // sum_{d>=1} x_d^2
  float x0 = xs[0];
  float u0 = x0 + fminf(1.f - x0, -EPSF) - 1.f;
  float scale = acoshf(fmaxf(x0, 1.f + EPSF)) /
                sqrtf(fmaxf(sx - u0 * u0, EPSF));

  // ---- mu_k = scale * sum_{d>=1} x_d W[k][d] -------------------------------
  const float* Wr = W + k * 128;
  float acc = 0.f;
  #pragma unroll 4
  for (int d = 1; d < 128; ++d) acc += xs[d] * Wr[d];
  float p = (k == 0) ? 0.f : scale * acc;                // proj_tan -> p0 = 0

  // ---- h = hproj(expmap(o_out, p)) ----------------------------------------
  float R1 = blk_reduce64(p * p, red);
  float th = fmaxf(sqrtf(fmaxf(R1, EPSF)), MINNF);
  float s  = sinhf(th) / th;
  float hk = s * p;                                      // spatial (0 at k==0)
  float R2 = blk_reduce64(hk * hk, red);
  float h0 = sqrtf(1.f + R2);

  // ---- pt = ptransp(o_out, h, b) ------------------------------------------
  float u0h  = h0 + fminf(1.f - h0, -EPSF) - 1.f;
  float dist = acoshf(fmaxf(h0, 1.f + EPSF));
  float g1   = dist / sqrtf(fmaxf(R2 - u0h * u0h, EPSF));      // logmap(o,h)
  float xy2  = fminf(1.f - h0, -EPSF) - 1.f;                   // logmap(h,o)
  float u02  = 1.f + xy2 * h0;
  float g2   = dist * xy2 / sqrtf(fmaxf(xy2 * xy2 * R2 - u02 * u02, EPSF));
  float d2   = fmaxf(dist * dist, MINNF);
  float bk   = b[k];
  float R3   = blk_reduce64((k == 0) ? 0.f : hk * bk, red);
  float alpha = g1 * R3 / d2;
  float wk   = (k == 0) ? 0.f : bk - alpha * (g1 + g2) * hk;   // spatial of pt
  float R4   = blk_reduce64(hk * wk, red);
  float pt0  = R4 / fmaxf(h0, MINNF);

  // ---- h = expmap(h, pt) ---------------------------------------------------
  float R5  = blk_reduce64(wk * wk, red);
  float th2 = fmaxf(sqrtf(fmaxf(R5 - pt0 * pt0, EPSF)), MINNF);
  float ch  = coshf(th2), sh = sinhf(th2) / th2;
  float hnk = ch * hk + sh * wk;                         // spatial
  float R6  = blk_reduce64(hnk * hnk, red);
  float hn0 = sqrtf(1.f + R6);

  // ---- h_tan = logmap(o_out, h); attention scores --------------------------
  float u0t = hn0 + fminf(1.f - hn0, -EPSF) - 1.f;
  float gt  = acoshf(fmaxf(hn0, 1.f + EPSF)) /
              sqrtf(fmaxf(R6 - u0t * u0t, EPSF));
  float htk = gt * hnk;                                  // h_tan spatial (0 @ k==0)
  float R7  = blk_reduce64(htk * aw[k], red);
  float R8  = blk_reduce64(htk * aw[64 + k], red);

  if (k == 0) {
    H[i * 64] = hn0;
    Q[i]  = R6 - hn0 * hn0;                              // mdot(h,h) ~ -1
    SL[i] = R7;
    SR[i] = R8;
  } else {
    H[i * 64 + k] = hnk;
  }
}

// ---------------------------------------------------------------------------
// Kernel 3: pairwise phase. Block = 1 wave (32 threads).
//  blockIdx.x -> i-tile (16 rows), blockIdx.y -> j-chunk (128 cols = 8 tiles).
//  GEMM1 (WMMA f32 16x16x4): m_ij = Minkowski Gram (time col of A negated).
//  Per-pair scalars: a = adj*sigmoid, xy, uu = q_j + 2*xy*m + xy^2*q_i,
//                    c = a*acosh(max(-m,1+eps))/sqrt(max(uu,eps)).
//  GEMM2 (WMMA): Spart += C(16x16) x H(16x64), c re-laid out via LDS.
// ---------------------------------------------------------------------------
__global__ __launch_bounds__(32)
void hgcn_pairB(const float* __restrict__ H, const float* __restrict__ Q,
                const float* __restrict__ SL, const float* __restrict__ SR,
                const float* __restrict__ adj, const float* __restrict__ abp,
                float* __restrict__ Spart, float* __restrict__ Wpart) {
  __shared__ float cl[16][17];
  __shared__ float wl[16][17];
  __shared__ float qi_s[16];
  __shared__ float sl_s[16];

  const int i0 = blockIdx.x * 16;
  const int jc = blockIdx.y;
  const int t  = threadIdx.x;
  const int lo = (t < 16) ? 1 : 0;
  const int r  = t & 15;
  const float ab = abp[0];

  if (t < 16) { qi_s[t] = Q[i0 + t]; sl_s[t] = SL[i0 + t]; }
  __syncthreads();

  // Preload A fragments for the Gram GEMM (invariant over j). A(16 x K=64),
  // lanes 0-15: K%4 in {0,1}; lanes 16-31: K%4 in {2,3}. Negate time column.
  float aX[16], aY[16];
  const float* Hi = H + (size_t)(i0 + r) * 64;
  #pragma unroll
  for (int kk = 0; kk < 16; ++kk) {
    int c0 = kk * 4 + (lo ? 0 : 2);
    float vx = Hi[c0], vy = Hi[c0 + 1];
    if (c0 == 0) vx = -vx;                               // Minkowski metric
    aX[kk] = vx; aY[kk] = vy;
  }

  const v8f vzero = {0.f, 0.f, 0.f, 0.f, 0.f, 0.f, 0.f, 0.f};
  v8f Sacc[4];
  #pragma unroll
  for (int cc = 0; cc < 4; ++cc) Sacc[cc] = vzero;
  float wpriv[8] = {0.f, 0.f, 0.f, 0.f, 0.f, 0.f, 0.f, 0.f};

  for (int jt = 0; jt < 8; ++jt) {
    const int j0 = jc * 128 + jt * 16;
    const float* Hj = H + (size_t)(j0 + r) * 64;

    // ---- Gram tile: m = A(16x64) x B(64x16), K-steps of 4 ------------------
    v8f m = vzero;
    #pragma unroll
    for (int kk = 0; kk < 16; ++kk) {
      int c0 = kk * 4 + (lo ? 0 : 2);
      v2f a = {aX[kk], aY[kk]};
      v2f bb = {Hj[c0], Hj[c0 + 1]};
      m = __builtin_amdgcn_wmma_f32_16x16x4_f32(false, a, false, bb,
                                                (short)0, m, false, false);
    }

    float qj = Q[j0 + r];
    float zr = SR[j0 + r];

    __syncthreads();   // cl from previous iteration fully consumed
    #pragma unroll
    for (int p = 0; p < 8; ++p) {
      int il = p + (lo ? 0 : 8);                         // local i row
      float mm = m[p];
      float sg = 1.f / (1.f + expf(-(sl_s[il] + zr + ab)));
      float aa = adj[(size_t)(i0 + il) * 1024 + (j0 + r)] * sg;
      float xy = fminf(mm + 1.f, -EPSF) - 1.f;
      float uu = qj + 2.f * xy * mm + xy * xy * qi_s[il];
      float nu = sqrtf(fmaxf(uu, EPSF));
      float dd = acoshf(fmaxf(-mm, 1.f + EPSF));
      float c  = aa * dd / nu;
      cl[il][r] = c;
      wpriv[p] += c * xy;
    }
    __syncthreads();

    // ---- Support tile: Sacc += C(16x16) x H(16x64) -------------------------
    #pragma unroll
    for (int cc = 0; cc < 4; ++cc) {
      #pragma unroll
      for (int k4 = 0; k4 < 4; ++k4) {
        int kb = k4 * 4 + (lo ? 0 : 2);
        v2f ca = {cl[r][kb], cl[r][kb + 1]};
        v2f cb = {H[(size_t)(j0 + kb) * 64 + cc * 16 + r],
                  H[(size_t)(j0 + kb + 1) * 64 + cc * 16 + r]};
        Sacc[cc] = __builtin_amdgcn_wmma_f32_16x16x4_f32(false, ca, false, cb,
                                                         (short)0, Sacc[cc],
                                                         false, false);
      }
    }
  }

  // ---- write partial support + partial wsum -------------------------------
  #pragma unroll
  for (int cc = 0; cc < 4; ++cc)
    #pragma unroll
    for (int p = 0; p < 8; ++p) {
      int M = p + (lo ? 0 : 8);
      Spart[((size_t)jc * 1024 + i0 + M) * 64 + cc * 16 + r] = Sacc[cc][p];
    }

  #pragma unroll
  for (int p = 0; p < 8; ++p) wl[p + (lo ? 0 : 8)][r] = wpriv[p];
  __syncthreads();
  if (t < 16) {
    float sum = 0.f;
    #pragma unroll
    for (int n2 = 0; n2 < 16; ++n2) sum += wl[t][n2];
    Wpart[jc * 1024 + i0 + t] = sum;
  }
}

// ---------------------------------------------------------------------------
// Kernel 4: reduce j-chunk partials (fixed order -> deterministic) + phase C:
//  S_k = Sg_k + wsum*h_k (k>=1);  S0 = (h_spatial . S_spatial)/h0
//  out = hproj(expmap(o, relu(logmap(o, hproj(expmap(h, S))))))
// ---------------------------------------------------------------------------
__global__ __launch_bounds__(64)
void hgcn_finalC(const float* __restrict__ H, const float* __restrict__ Spart,
                 const float* __restrict__ Wpart, float* __restrict__ out) {
  __shared__ float red[64];
  const int i = blockIdx.x;
  const int k = threadIdx.x;

  float hn  = H[i * 64 + k];
  float hn0 = H[i * 64];
  float hk  = (k == 0) ? 0.f : hn;

  float Sg = 0.f, wsum = 0.f;
  #pragma unroll
  for (int c = 0; c < 8; ++c) {
    Sg   += Spart[((size_t)c * 1024 + i) * 64 + k];
    wsum += Wpart[c * 1024 + i];
  }
  float Sk = (k == 0) ? 0.f : Sg + wsum * hk;

  float R1 = blk_reduce64(hk * Sk, red);
  float S0 = R1 / fmaxf(hn0, MINNF);

  // expmap(h, S) + hproj
  float R2 = blk_reduce64(Sk * Sk, red);
  float th = fmaxf(sqrtf(fmaxf(R2 - S0 * S0, EPSF)), MINNF);
  float ch = coshf(th), sh = sinhf(th) / th;
  float ek = ch * hk + sh * Sk;                          // spatial
  float R3 = blk_reduce64(ek * ek, red);
  float e0 = sqrtf(1.f + R3);

  // relu(logmap(o, out)); proj_tan at o keeps time comp 0
  float u0 = e0 + fminf(1.f - e0, -EPSF) - 1.f;
  float g  = acoshf(fmaxf(e0, 1.f + EPSF)) /
             sqrtf(fmaxf(R3 - u0 * u0, EPSF));
  float xt = fmaxf(g * ek, 0.f);                         // 0 at k==0

  // hproj(expmap(o, xt))
  float R4  = blk_reduce64(xt * xt, red);
  float th2 = fmaxf(sqrtf(fmaxf(R4, EPSF)), MINNF);
  float s2  = sinhf(th2) / th2;
  float fk  = s2 * xt;
  float R5  = blk_reduce64(fk * fk, red);
  out[i * 64 + k] = (k == 0) ? sqrtf(1.f + R5) : fk;
}

// ---------------------------------------------------------------------------
extern "C" void kernel_launch(void* const* d_in, const int* in_sizes, int n_in,
                              void* d_out, int out_size, void* d_ws, size_t ws_size,
                              hipStream_t stream) {
  const float*     x  = (const float*)d_in[0];
  const long long* ei = (const long long*)d_in[1];
  const float*     W  = (const float*)d_in[2];
  const float*     b  = (const float*)d_in[3];
  const float*     aw = (const float*)d_in[4];
  const float*     ab = (const float*)d_in[5];
  float* out = (float*)d_out;

  const int E = in_sizes[1] / 2;                         // 65536

  // workspace layout (floats): adj | H | Q | SL | SR | Spart | Wpart
  float* ws    = (float*)d_ws;
  float* adj   = ws;                                     // 1024*1024
  float* H     = adj   + 1024 * 1024;                    // 1024*64
  float* Q     = H     + 1024 * 64;                      // 1024
  float* SL    = Q     + 1024;                           // 1024
  float* SR    = SL    + 1024;                           // 1024
  float* Spart = SR    + 1024;                           // 8*1024*64
  float* Wpart = Spart + 8 * 1024 * 64;                  // 8*1024

  hipMemsetAsync(adj, 0, (size_t)1024 * 1024 * sizeof(float), stream);
  hgcn_edge_scatter<<<(E + 255) / 256, 256, 0, stream>>>(ei, adj, E);
  hgcn_nodeA<<<1024, 64, 0, stream>>>(x, W, b, aw, H, Q, SL, SR);
  hgcn_pairB<<<dim3(64, 8), 32, 0, stream>>>(H, Q, SL, SR, adj, ab, Spart, Wpart);
  hgcn_finalC<<<1024, 64, 0, stream>>>(H, Spart, Wpart, out);
}